// SoftPatternClassifier_1649267442164
// MI455X (gfx1250) — compile-verified
//
#include <hip/hip_runtime.h>
#include <hip/hip_bf16.h>

// Problem constants from the reference
#define BB  256
#define TT  128
#define PP  200
#define LM1 5
#define DD  300
#define VV  10000
#define CC  2
#define NN  (PP * LM1)   // 1000
#define NEGINF (-__builtin_inff())
#define POSINF (__builtin_inff())

typedef float v2f __attribute__((ext_vector_type(2)));
typedef float v8f __attribute__((ext_vector_type(8)));

// ---------------------------------------------------------------------------
// Kernel 1: table[v][pl] = sum_d diags[pl][d] * emb[d][v] + bias[pl]
// M = V = 10000 (625 tiles of 16), N = 1000, K = 300.
// One wave computes a 16x64 strip: 4 adjacent 16x16 N-tiles sharing one A
// operand per k-step -> 2 A loads + 4 B loads feed 4 v_wmma_f32_16x16x4_f32.
// ---------------------------------------------------------------------------
__global__ __launch_bounds__(256) void table_gemm_kernel(
    const float* __restrict__ emb,    // (D, V) row-major
    const float* __restrict__ diags,  // (N, D) row-major
    const float* __restrict__ bias,   // (N,)
    float* __restrict__ table)        // (V, N) row-major
{
    const int NG = 16;                           // n-groups (4 tiles = 64 cols each)
    const int TV = 625;                          // v-tiles
    int wave = blockIdx.x * (blockDim.x >> 5) + (threadIdx.x >> 5);
    int lane = threadIdx.x & 31;
    int tv = wave / NG;
    int g  = wave - tv * NG;
    if (tv >= TV) return;

    int v0 = tv * 16;
    int ml = lane & 15;     // A: m index; B/C/D: n index
    int hi = lane >> 4;     // 0 or 1 -> K pair / M half selector

    int acol = v0 + ml;                          // v column for A loads
    const float* aptr = emb + acol;              // + d*V

    // Per-subtile B row (pl index), clamped at the N edge; bias into acc init.
    int brow[4];
    const float* bptr[4];
    v8f acc[4];
#pragma unroll
    for (int s = 0; s < 4; ++s) {
        brow[s] = g * 64 + s * 16 + ml;
        int bc  = brow[s] < NN ? brow[s] : (NN - 1);
        bptr[s] = diags + (size_t)bc * DD;
        float bv = bias[bc];
#pragma unroll
        for (int i = 0; i < 8; ++i) acc[s][i] = bv;
    }

#if __has_builtin(__builtin_amdgcn_wmma_f32_16x16x4_f32)
#pragma unroll 5
    for (int k = 0; k < DD; k += 4) {
        int ka = k + hi * 2;
        v2f A;
        // 32-bit A 16x4 layout: lane = m + 16*(k>=2), vgpr = k&1
        A.x = aptr[(size_t)ka * VV];
        A.y = aptr[(size_t)(ka + 1) * VV];
#pragma unroll
        for (int s = 0; s < 4; ++s) {
            // B 4x16 mirrors A with n in place of m; d,d+1 contiguous -> b64
            v2f B = *(const v2f*)(bptr[s] + ka);
            acc[s] = __builtin_amdgcn_wmma_f32_16x16x4_f32(
                /*neg_a=*/false, A, /*neg_b=*/false, B,
                /*c_mod=*/(short)0, acc[s], /*reuse_a=*/false, /*reuse_b=*/false);
        }
    }
#else
    // Scalar fallback: acc[s][r] -> (m = v0+hi*8+r, n = g*64+s*16+ml)
    for (int d = 0; d < DD; ++d) {
#pragma unroll
        for (int s = 0; s < 4; ++s) {
            float bvv = bptr[s][d];
#pragma unroll
            for (int r = 0; r < 8; ++r)
                acc[s][r] = fmaf(emb[(size_t)d * VV + (v0 + hi * 8 + r)], bvv,
                                 acc[s][r]);
        }
    }
#endif

    // D layout: vgpr r = row (r + 8*hi), col = lane&15
#pragma unroll
    for (int s = 0; s < 4; ++s) {
        if (brow[s] < NN) {
            float* outp = table + (size_t)(v0 + hi * 8) * NN + brow[s];
#pragma unroll
            for (int r = 0; r < 8; ++r)
                outp[(size_t)r * NN] = acc[s][r];
        }
    }
}

// ---------------------------------------------------------------------------
// Kernel 2: per-document max-plus scan + layernorm-binarize + linear head.
// One block (256 threads) per document; thread p < 200 owns pattern p.
// ---------------------------------------------------------------------------
template <bool IS_MIN>
__device__ __forceinline__ float block_reduce(float* red, int tid, float v)
{
    red[tid] = v;
    __syncthreads();
#pragma unroll
    for (int off = 128; off > 0; off >>= 1) {
        if (tid < off)
            red[tid] = IS_MIN ? fminf(red[tid], red[tid + off])
                              : (red[tid] + red[tid + off]);
        __syncthreads();
    }
    float r = red[0];
    __syncthreads();
    return r;
}

__global__ __launch_bounds__(256) void scan_finalize_kernel(
    const int*   __restrict__ docs,       // (B, T)
    const int*   __restrict__ doc_lens,   // (B,)
    const int*   __restrict__ end_states, // (P,) values in {2..5}
    const float* __restrict__ wildcards,  // (P, LM1)
    const float* __restrict__ lin_w,      // (C, P)
    const float* __restrict__ lin_b,      // (C,)
    const float* __restrict__ table,      // (V, N)
    float*       __restrict__ out)        // (B, C)
{
    __shared__ float row[NN];    // one table row (4 KB)
    __shared__ float red[256];

    const int b   = blockIdx.x;
    const int tid = threadIdx.x;
    const int* doc = docs + b * TT;
    const int  L   = doc_lens[b];
    const bool own = tid < PP;

    float wc[LM1];
    int eidx = 2;
    if (own) {
#pragma unroll
        for (int j = 0; j < LM1; ++j) wc[j] = wildcards[tid * LM1 + j];
        eidx = end_states[tid];
    }

    float hid[LM1 + 1];
    hid[0] = 0.0f;
#pragma unroll
    for (int j = 1; j <= LM1; ++j) hid[j] = NEGINF;
    float sc = NEGINF;

    const int base = tid * LM1;
    for (int t = 0; t < L; ++t) {
        int tok = doc[t];   // uniform across block -> scalar load
        __syncthreads();
        // stage 4 KB row: 250 float4 loads (row stride 4000 B is 16-aligned)
        if (tid < NN / 4) {
            const float4* src = (const float4*)(table + (size_t)tok * NN);
            ((float4*)row)[tid] = src[tid];
        }
        __syncthreads();
        if (own) {
            // hid[j] = hid[j-1] + max(tr[j-1], wc[j-1]); descending keeps prev
#pragma unroll
            for (int j = LM1; j >= 1; --j)
                hid[j] = hid[j - 1] + fmaxf(row[base + j - 1], wc[j - 1]);
            float endv = (eidx == 2) ? hid[2]
                       : (eidx == 3) ? hid[3]
                       : (eidx == 4) ? hid[4] : hid[5];
            sc = fmaxf(sc, endv);
        }
    }

    // scores == -inf -> +inf, take row min, replace +inf with row min
    float s_for_min = own ? ((sc == NEGINF) ? POSINF : sc) : POSINF;
    float rmin = block_reduce<true>(red, tid, s_for_min);
    float s = own ? ((sc == NEGINF) ? rmin : sc) : 0.0f;

    float sum   = block_reduce<false>(red, tid, s);
    float sumsq = block_reduce<false>(red, tid, s * s);
    float mu  = sum / (float)PP;
    float var = sumsq / (float)PP - mu * mu;
    (void)var; // (scores-mu)/sqrt(var+eps) > 0  <=>  scores > mu

    float bin = (own && (s > mu)) ? 1.0f : 0.0f;
    float d0 = block_reduce<false>(red, tid, own ? bin * lin_w[tid]       : 0.0f);
    float d1 = block_reduce<false>(red, tid, own ? bin * lin_w[PP + tid]  : 0.0f);

    if (tid == 0) {
        out[b * CC + 0] = d0 + lin_b[0];
        out[b * CC + 1] = d1 + lin_b[1];
    }
}

// ---------------------------------------------------------------------------
extern "C" void kernel_launch(void* const* d_in, const int* in_sizes, int n_in,
                              void* d_out, int out_size, void* d_ws, size_t ws_size,
                              hipStream_t stream)
{
    (void)in_sizes; (void)n_in; (void)out_size; (void)ws_size;
    const int*   docs       = (const int*)d_in[0];
    const int*   doc_lens   = (const int*)d_in[1];
    const int*   end_states = (const int*)d_in[2];
    const float* emb        = (const float*)d_in[3];
    const float* diags      = (const float*)d_in[4];
    const float* bias       = (const float*)d_in[5];
    const float* wildcards  = (const float*)d_in[6];
    const float* lin_w      = (const float*)d_in[7];
    const float* lin_b      = (const float*)d_in[8];
    float* out   = (float*)d_out;
    float* table = (float*)d_ws;    // 10000*1000*4 = 40 MB scratch

    const int waves  = 625 * 16;                 // v-tiles x n-groups
    const int wavesPerBlock = 8;                 // 256 threads, wave32
    const int blocks = (waves + wavesPerBlock - 1) / wavesPerBlock;
    table_gemm_kernel<<<blocks, 256, 0, stream>>>(emb, diags, bias, table);

    scan_finalize_kernel<<<BB, 256, 0, stream>>>(docs, doc_lens, end_states,
                                                 wildcards, lin_w, lin_b,
                                                 table, out);
}